// GNN_64570538328069
// MI455X (gfx1250) — compile-verified
//
#include <hip/hip_runtime.h>

typedef float v2f __attribute__((ext_vector_type(2)));
typedef float v8f __attribute__((ext_vector_type(8)));

#define D 128            // feature dim (both layers)
#define LN_EPS 1e-5f
#define XT_STRIDE 132    // padded LDS row stride (floats): bank-spread + 16B rows

// ---------------------------------------------------------------------------
// Stage a 16xD fp32 tile G[row0:row0+16, :] into LDS via async global->LDS DMA
// (GLOBAL_LOAD_ASYNC_TO_LDS_B128, tracked by ASYNCcnt). 256 threads issue
// 512 x b128 transfers (2 per thread). Caller must s_wait_asynccnt + barrier.
// ---------------------------------------------------------------------------
__device__ __forceinline__ void async_stage_tile(const float* __restrict__ G,
                                                 int row0, float* xt) {
    const unsigned lds0 = (unsigned)(size_t)xt;   // low 32 bits == LDS offset
    const int t = threadIdx.x;
#pragma unroll
    for (int i = 0; i < 2; ++i) {
        const int idx = t + i * 256;              // 0..511
        const int row = idx >> 5;                 // 16 rows
        const int c4  = idx & 31;                 // 32 float4 per row
        const float* gp = G + (size_t)(row0 + row) * D + c4 * 4;
        const unsigned laddr = lds0 + (unsigned)(row * XT_STRIDE + c4 * 4) * 4u;
        asm volatile("global_load_async_to_lds_b128 %0, %1, off"
                     :: "v"(laddr), "v"((unsigned long long)(size_t)gp)
                     : "memory");
    }
}

__device__ __forceinline__ void wait_async_and_sync() {
    asm volatile("s_wait_asynccnt 0x0" ::: "memory");
    __syncthreads();
}

// ---------------------------------------------------------------------------
// One wave computes a 16x16 fp32 C-tile:  XT(lds) @ W[:, col0:col0+16], K=128,
// via 32 chained V_WMMA_F32_16X16X4_F32.
// A lane layout (16x4 MxK): lane L: M = L%16, VGPR0 -> K = 2*(L/16), VGPR1 -> K+1
// B lane layout (4x16 KxN): lane L: N = L%16, VGPR0 -> K = 2*(L/16), VGPR1 -> K+1
// C/D layout: VGPR r -> (M = r + 8*(L/16), N = L%16)
// ---------------------------------------------------------------------------
__device__ __forceinline__ v8f wmma_tile_lds(const float* xt,
                                             const float* __restrict__ W,
                                             int col0, v8f acc) {
    const int lane = threadIdx.x & 31;
    const int half = lane >> 4;      // 0 or 1
    const int mod  = lane & 15;
    const float* xrow = xt + mod * XT_STRIDE;      // LDS, ds_load_b64 per step
#pragma unroll
    for (int k = 0; k < D; k += 4) {
        const int ka = k + 2 * half;
        v2f a, b;
        a.x = xrow[ka];
        a.y = xrow[ka + 1];
        b.x = W[(size_t)ka * D + col0 + mod];
        b.y = W[(size_t)(ka + 1) * D + col0 + mod];
        acc = __builtin_amdgcn_wmma_f32_16x16x4_f32(false, a, false, b,
                                                    (short)0, acc, false, false);
    }
    return acc;
}

// H = relu(X @ Wpool + bpool)    grid = N/16 blocks, 256 threads (8 waves)
__global__ void gemm_pool(const float* __restrict__ X, const float* __restrict__ W,
                          const float* __restrict__ bias, float* __restrict__ H) {
    __shared__ float xt[16 * XT_STRIDE];
    const int row0 = blockIdx.x * 16;
    async_stage_tile(X, row0, xt);
    wait_async_and_sync();

    const int wave = threadIdx.x >> 5;           // 0..7 -> N-tile
    const int col0 = wave * 16;
    v8f acc = {};
    acc = wmma_tile_lds(xt, W, col0, acc);

    const int lane = threadIdx.x & 31;
    const int half = lane >> 4;
    const int mod  = lane & 15;
    const float b = bias[col0 + mod];
    union { v8f v; float f[8]; } u; u.v = acc;
#pragma unroll
    for (int r = 0; r < 8; ++r) {
        float v = u.f[r] + b;
        v = v > 0.f ? v : 0.f;
        H[(size_t)(row0 + r + 8 * half) * D + col0 + mod] = v;
    }
}

// Y = X @ Wself + NEI @ Wneigh + bias   (no activation; LN comes after)
__global__ void gemm_dual(const float* __restrict__ X, const float* __restrict__ Wself,
                          const float* __restrict__ NEI, const float* __restrict__ Wneigh,
                          const float* __restrict__ bias, float* __restrict__ Y) {
    __shared__ float xt[2][16 * XT_STRIDE];
    const int row0 = blockIdx.x * 16;
    async_stage_tile(X,   row0, xt[0]);
    async_stage_tile(NEI, row0, xt[1]);
    wait_async_and_sync();

    const int wave = threadIdx.x >> 5;
    const int col0 = wave * 16;
    v8f acc = {};
    acc = wmma_tile_lds(xt[0], Wself,  col0, acc);
    acc = wmma_tile_lds(xt[1], Wneigh, col0, acc);

    const int lane = threadIdx.x & 31;
    const int half = lane >> 4;
    const int mod  = lane & 15;
    const float b = bias[col0 + mod];
    union { v8f v; float f[8]; } u; u.v = acc;
#pragma unroll
    for (int r = 0; r < 8; ++r)
        Y[(size_t)(row0 + r + 8 * half) * D + col0 + mod] = u.f[r] + b;
}

__global__ void zero_f32(float* __restrict__ p, int n) {
    int i = blockIdx.x * blockDim.x + threadIdx.x;
    if (i < n) p[i] = 0.f;
}

// Wave-per-edge segment-max scatter. H rows are ReLU outputs (>= 0), NEIGH is
// zero-initialized, so unsigned-int atomic max == float max and isolated nodes
// correctly stay 0 (DGL semantics).
__global__ void edge_max(const float* __restrict__ H, const int* __restrict__ src,
                         const int* __restrict__ dst, float* __restrict__ NEIGH,
                         int n_edges) {
    const int lane = threadIdx.x & 31;
    const int e = blockIdx.x * (blockDim.x >> 5) + (threadIdx.x >> 5);
    if (e >= n_edges) return;
    const int s = src[e];
    const int d = dst[e];
    const float4 v = ((const float4*)(H + (size_t)s * D))[lane];     // 512B/wave gather
    unsigned* np = (unsigned*)(NEIGH + (size_t)d * D) + lane * 4;
    atomicMax(np + 0, __float_as_uint(v.x));
    atomicMax(np + 1, __float_as_uint(v.y));
    atomicMax(np + 2, __float_as_uint(v.z));
    atomicMax(np + 3, __float_as_uint(v.w));
}

// Out = relu(layernorm(Y) * gamma + beta), wave per row (8 rows / 256-thr block)
__global__ void ln_relu(const float* __restrict__ Y, const float* __restrict__ gamma,
                        const float* __restrict__ beta, float* __restrict__ Out,
                        int n_rows) {
    const int lane = threadIdx.x & 31;
    const int row = blockIdx.x * (blockDim.x >> 5) + (threadIdx.x >> 5);
    if (row >= n_rows) return;
    const float4 v = ((const float4*)(Y + (size_t)row * D))[lane];
    float s = v.x + v.y + v.z + v.w;
    float q = v.x * v.x + v.y * v.y + v.z * v.z + v.w * v.w;
#pragma unroll
    for (int off = 16; off > 0; off >>= 1) {         // wave32 reduction
        s += __shfl_xor(s, off, 32);
        q += __shfl_xor(q, off, 32);
    }
    const float mu  = s * (1.f / D);
    const float var = q * (1.f / D) - mu * mu;
    const float inv = rsqrtf(var + LN_EPS);
    const float4 g4 = ((const float4*)gamma)[lane];
    const float4 b4 = ((const float4*)beta)[lane];
    float4 o;
    o.x = fmaxf((v.x - mu) * inv * g4.x + b4.x, 0.f);
    o.y = fmaxf((v.y - mu) * inv * g4.y + b4.y, 0.f);
    o.z = fmaxf((v.z - mu) * inv * g4.z + b4.z, 0.f);
    o.w = fmaxf((v.w - mu) * inv * g4.w + b4.w, 0.f);
    ((float4*)(Out + (size_t)row * D))[lane] = o;
}

// Column-wise max over all rows (X >= 0 post-ReLU; g pre-zeroed; uint trick).
__global__ void col_max(const float* __restrict__ X, float* __restrict__ g, int n_rows) {
    const int col = threadIdx.x;                      // 128 threads
    float m = 0.f;
    for (int r = blockIdx.x; r < n_rows; r += gridDim.x)
        m = fmaxf(m, X[(size_t)r * D + col]);
    atomicMax((unsigned*)g + col, __float_as_uint(m));
}

// out = (g @ Wfc1 + bfc1) @ Wfc + bfc   (single block, 128 threads)
__global__ void head(const float* __restrict__ g, const float* __restrict__ Wfc1,
                     const float* __restrict__ bfc1, const float* __restrict__ Wfc,
                     const float* __restrict__ bfc, float* __restrict__ out) {
    __shared__ float red[128];
    const int j = threadIdx.x;
    float t = bfc1[j];
    for (int k = 0; k < D; ++k) t += g[k] * Wfc1[(size_t)k * D + j];
    red[j] = t * Wfc[j];
    __syncthreads();
    for (int s = 64; s > 0; s >>= 1) {
        if (j < s) red[j] += red[j + s];
        __syncthreads();
    }
    if (j == 0) out[0] = red[0] + bfc[0];
}

extern "C" void kernel_launch(void* const* d_in, const int* in_sizes, int n_in,
                              void* d_out, int out_size, void* d_ws, size_t ws_size,
                              hipStream_t stream) {
    const float* features = (const float*)d_in[0];
    const int*   src      = (const int*)d_in[1];
    const int*   dst      = (const int*)d_in[2];
    const float* Wpool0   = (const float*)d_in[3];
    const float* bpool0   = (const float*)d_in[4];
    const float* Wself0   = (const float*)d_in[5];
    const float* Wneigh0  = (const float*)d_in[6];
    const float* bias0    = (const float*)d_in[7];
    const float* gamma0   = (const float*)d_in[8];
    const float* beta0    = (const float*)d_in[9];
    const float* Wpool1   = (const float*)d_in[10];
    const float* bpool1   = (const float*)d_in[11];
    const float* Wself1   = (const float*)d_in[12];
    const float* Wneigh1  = (const float*)d_in[13];
    const float* bias1    = (const float*)d_in[14];
    const float* gamma1   = (const float*)d_in[15];
    const float* beta1    = (const float*)d_in[16];
    const float* Wfc1     = (const float*)d_in[17];
    const float* bfc1     = (const float*)d_in[18];
    const float* Wfc      = (const float*)d_in[19];
    const float* bfc      = (const float*)d_in[20];
    float* out = (float*)d_out;

    const int N = in_sizes[0] / D;     // 50000
    const int E = in_sizes[1];         // 800000

    const size_t A = (size_t)N * D * sizeof(float);     // 25.6 MB
    char* ws = (char*)d_ws;
    float* bufH = (float*)(ws);            // h, later reused for pre-LN Y
    float* bufN = (float*)(ws + A);        // neigh (pooled)
    float* bufX = (float*)(ws + 2 * A);    // layer-1 output x1
    float* gbuf = (float*)(ws + 3 * A);    // 128-float readout max

    const int gemmBlocks = N / 16;                       // 3125
    const int edgeBlocks = (E + 7) / 8;                  // 8 edges (waves) / block
    const int zeroBlocks = (N * D + 255) / 256;
    const int lnBlocks   = (N + 7) / 8;

    // ---- layer 0 ----
    gemm_pool<<<gemmBlocks, 256, 0, stream>>>(features, Wpool0, bpool0, bufH);
    zero_f32<<<zeroBlocks, 256, 0, stream>>>(bufN, N * D);
    edge_max<<<edgeBlocks, 256, 0, stream>>>(bufH, src, dst, bufN, E);
    gemm_dual<<<gemmBlocks, 256, 0, stream>>>(features, Wself0, bufN, Wneigh0, bias0, bufH);
    ln_relu<<<lnBlocks, 256, 0, stream>>>(bufH, gamma0, beta0, bufX, N);

    // ---- layer 1 ----
    gemm_pool<<<gemmBlocks, 256, 0, stream>>>(bufX, Wpool1, bpool1, bufH);
    zero_f32<<<zeroBlocks, 256, 0, stream>>>(bufN, N * D);
    edge_max<<<edgeBlocks, 256, 0, stream>>>(bufH, src, dst, bufN, E);
    gemm_dual<<<gemmBlocks, 256, 0, stream>>>(bufX, Wself1, bufN, Wneigh1, bias1, bufH);
    ln_relu<<<lnBlocks, 256, 0, stream>>>(bufH, gamma1, beta1, bufH, N);   // in-place

    // ---- readout + head ----
    zero_f32<<<1, 128, 0, stream>>>(gbuf, 128);
    col_max<<<512, 128, 0, stream>>>(bufH, gbuf, N);
    head<<<1, 128, 0, stream>>>(gbuf, Wfc1, bfc1, Wfc, bfc, out);
}